// EarthMoverDistance_19980187861537
// MI455X (gfx1250) — compile-verified
//
#include <hip/hip_runtime.h>
#include <math.h>

#define N     1024
#define NP1   1025
#define BLOCK 256
#define INFV  1e9f

typedef __attribute__((ext_vector_type(2))) float v2f;
typedef __attribute__((ext_vector_type(8))) float v8f;

// ---------------------------------------------------------------------------
// Kernel 1: cost[b][m][n] = || S1[b,m,:] - S2[b,n,:] ||_2  via WMMA.
// dot = S1 . S2^T with V_WMMA_F32_16X16X4_F32 (K=3 padded to 4),
// cost = sqrt(max(0, |s1|^2 + |s2|^2 - 2*dot)).
// One wave per 16x16 tile; grid = (N/16, N/16, B).
// ---------------------------------------------------------------------------
__global__ __launch_bounds__(32)
void emd_cost_wmma(const float* __restrict__ S1, const float* __restrict__ S2,
                   float* __restrict__ cost)
{
    const int lane = threadIdx.x;        // 0..31, EXEC all-ones (no divergence)
    const int n0   = blockIdx.x * 16;
    const int m0   = blockIdx.y * 16;
    const int b    = blockIdx.z;
    const float* s1 = S1 + (size_t)b * N * 3;
    const float* s2 = S2 + (size_t)b * N * 3;

    const int half = lane >> 4;          // 0 or 1
    const int l16  = lane & 15;

    // A fragment (16x4, MxK): VGPR r, lane L -> m = L%16, k = (L/16)*2 + r
    // B fragment (4x16, KxN): VGPR r, lane L -> n = L%16, k = (L/16)*2 + r
    v2f afrag, bfrag;
    {
        const int m  = m0 + l16;
        const int n  = n0 + l16;
        const int k0 = half * 2;
        afrag.x = (k0 + 0 < 3) ? s1[m * 3 + k0 + 0] : 0.0f;
        afrag.y = (k0 + 1 < 3) ? s1[m * 3 + k0 + 1] : 0.0f;
        bfrag.x = (k0 + 0 < 3) ? s2[n * 3 + k0 + 0] : 0.0f;
        bfrag.y = (k0 + 1 < 3) ? s2[n * 3 + k0 + 1] : 0.0f;
    }
    v8f c = {};
    c = __builtin_amdgcn_wmma_f32_16x16x4_f32(false, afrag, false, bfrag,
                                              (short)0, c, false, false);

    // |s2[n]|^2 for this lane's column
    const int n  = n0 + l16;
    const float nx = s2[n * 3 + 0], ny = s2[n * 3 + 1], nz = s2[n * 3 + 2];
    const float n2 = nx * nx + ny * ny + nz * nz;

#pragma unroll
    for (int vv = 0; vv < 8; ++vv) {
        const int m = m0 + vv + half * 8;        // C/D layout: M = v (+8 for hi half)
        const float mx = s1[m * 3 + 0], my = s1[m * 3 + 1], mz = s1[m * 3 + 2];
        const float m2 = mx * mx + my * my + mz * mz;
        float d2 = m2 + n2 - 2.0f * c[vv];
        d2 = d2 > 0.0f ? d2 : 0.0f;
        cost[((size_t)b * N + m) * N + n] = sqrtf(d2);
    }
}

// ---------------------------------------------------------------------------
// Kernel 2: exact Jonker-Volgenant assignment, one workgroup per batch.
// All solver state lives in LDS (WGP has 320 KB). 256 threads = 8 wave32s.
// USE_COST=1: rows read from the (L2-resident) precomputed cost matrix.
// USE_COST=0: distances computed on the fly from LDS-resident point clouds.
// ---------------------------------------------------------------------------
__device__ __forceinline__ float wave_min_idx(float& val, int& idx)
{
#pragma unroll
    for (int off = 16; off > 0; off >>= 1) {
        float ov = __shfl_down(val, off, 32);
        int   oi = __shfl_down(idx, off, 32);
        if (ov < val) { val = ov; idx = oi; }
    }
    return val;
}

template <int USE_COST>
__global__ __launch_bounds__(BLOCK)
void emd_jv_solver(const float* __restrict__ S1, const float* __restrict__ S2,
                   const float* __restrict__ cost, float* __restrict__ batch_out)
{
    const int b   = blockIdx.x;
    const int tid = threadIdx.x;

    __shared__ float u[NP1], v[NP1], minv[NP1];
    __shared__ int   p[NP1], way[NP1], used[NP1];
    __shared__ float s1x[N], s1y[N], s1z[N], s2x[N], s2y[N], s2z[N];
    __shared__ float red_val[BLOCK / 32];
    __shared__ int   red_idx[BLOCK / 32];
    __shared__ int   s_j1;
    __shared__ float s_delta;

    if (!USE_COST) {
        for (int j = tid; j < N; j += BLOCK) {
            const float* q1 = S1 + ((size_t)b * N + j) * 3;
            const float* q2 = S2 + ((size_t)b * N + j) * 3;
            s1x[j] = q1[0]; s1y[j] = q1[1]; s1z[j] = q1[2];
            s2x[j] = q2[0]; s2y[j] = q2[1]; s2z[j] = q2[2];
        }
    }
    for (int j = tid; j < NP1; j += BLOCK) { u[j] = 0.0f; v[j] = 0.0f; p[j] = 0; }
    __syncthreads();

    const float* cbase = USE_COST ? (cost + (size_t)b * N * N) : nullptr;

    for (int i = 1; i <= N; ++i) {
        for (int j = tid; j < NP1; j += BLOCK) { minv[j] = INFV; used[j] = 0; way[j] = 0; }
        if (tid == 0) p[0] = i;
        __syncthreads();

        int j0 = 0;
        for (;;) {
            if (tid == 0) used[j0] = 1;
            __syncthreads();

            const int   i0  = p[j0];
            const float ui0 = u[i0];
            float r1x = 0.0f, r1y = 0.0f, r1z = 0.0f;
            const float* row = nullptr;
            if (USE_COST) {
                row = cbase + (size_t)(i0 - 1) * N;
            } else {
                r1x = s1x[i0 - 1]; r1y = s1y[i0 - 1]; r1z = s1z[i0 - 1];
            }

            // relax + local argmin over unused columns (striped, coalesced)
            float lmin = INFV; int lidx = 0;
            for (int j = 1 + tid; j <= N; j += BLOCK) {
                if (!used[j]) {
                    float aij;
                    if (USE_COST) {
                        aij = row[j - 1];
                    } else {
                        const float dx = r1x - s2x[j - 1];
                        const float dy = r1y - s2y[j - 1];
                        const float dz = r1z - s2z[j - 1];
                        aij = sqrtf(dx * dx + dy * dy + dz * dz);
                    }
                    const float cur = aij - ui0 - v[j];
                    if (cur < minv[j]) { minv[j] = cur; way[j] = j0; }
                    const float mj = minv[j];
                    if (mj < lmin) { lmin = mj; lidx = j; }
                }
            }
            wave_min_idx(lmin, lidx);
            if ((tid & 31) == 0) { red_val[tid >> 5] = lmin; red_idx[tid >> 5] = lidx; }
            __syncthreads();
            if (tid == 0) {
                float bm = red_val[0]; int bi = red_idx[0];
                for (int w = 1; w < BLOCK / 32; ++w)
                    if (red_val[w] < bm) { bm = red_val[w]; bi = red_idx[w]; }
                s_j1 = bi; s_delta = bm;
            }
            __syncthreads();
            const int   j1    = s_j1;
            const float delta = s_delta;

            // dual updates: used columns have pairwise-distinct p[j] -> race-free
            for (int j = tid; j < NP1; j += BLOCK) {
                if (used[j]) { u[p[j]] += delta; v[j] -= delta; }
                else         { minv[j] -= delta; }
            }
            __syncthreads();

            j0 = j1;
            if (p[j0] == 0) break;        // reached an unmatched column
        }

        // augment along alternating path (cheap LDS pointer chase)
        if (tid == 0) {
            int jj = j0;
            while (jj != 0) { const int jn = way[jj]; p[jj] = p[jn]; jj = jn; }
        }
        __syncthreads();
    }

    // total matched cost
    float local = 0.0f;
    for (int j = 1 + tid; j <= N; j += BLOCK) {
        const int i0 = p[j];
        if (USE_COST) {
            local += cbase[(size_t)(i0 - 1) * N + (j - 1)];
        } else {
            const float dx = s1x[i0 - 1] - s2x[j - 1];
            const float dy = s1y[i0 - 1] - s2y[j - 1];
            const float dz = s1z[i0 - 1] - s2z[j - 1];
            local += sqrtf(dx * dx + dy * dy + dz * dz);
        }
    }
#pragma unroll
    for (int off = 16; off > 0; off >>= 1) local += __shfl_down(local, off, 32);
    __syncthreads();
    if ((tid & 31) == 0) red_val[tid >> 5] = local;
    __syncthreads();
    if (tid == 0) {
        float s = 0.0f;
        for (int w = 0; w < BLOCK / 32; ++w) s += red_val[w];
        batch_out[b] = s / (float)N;
    }
}

// ---------------------------------------------------------------------------
// Kernel 3: deterministic scalar mean over batches (no float atomics).
// ---------------------------------------------------------------------------
__global__ void emd_finalize(const float* __restrict__ batch_out,
                             float* __restrict__ out, int B)
{
    float s = 0.0f;
    for (int b = 0; b < B; ++b) s += batch_out[b];
    out[0] = s / (float)B;
}

// ---------------------------------------------------------------------------
extern "C" void kernel_launch(void* const* d_in, const int* in_sizes, int n_in,
                              void* d_out, int out_size, void* d_ws, size_t ws_size,
                              hipStream_t stream)
{
    const float* S1 = (const float*)d_in[0];
    const float* S2 = (const float*)d_in[1];
    int B = in_sizes[0] / (N * 3);
    if (B < 1) B = 1;

    const size_t costBytes = (size_t)B * N * N * sizeof(float);
    const bool use_cost = (ws_size >= costBytes + 256);

    float* cost      = (float*)d_ws;
    float* batch_out = use_cost ? (float*)((char*)d_ws + costBytes) : (float*)d_ws;

    if (use_cost) {
        dim3 grid(N / 16, N / 16, B);
        emd_cost_wmma<<<grid, 32, 0, stream>>>(S1, S2, cost);
        emd_jv_solver<1><<<B, BLOCK, 0, stream>>>(S1, S2, cost, batch_out);
    } else {
        emd_jv_solver<0><<<B, BLOCK, 0, stream>>>(S1, S2, nullptr, batch_out);
    }
    emd_finalize<<<1, 1, 0, stream>>>(batch_out, (float*)d_out, B);
}